// DeattenuateLoss_69123203661888
// MI455X (gfx1250) — compile-verified
//
#include <hip/hip_runtime.h>
#include <hip/hip_bf16.h>

#define HH 1024
#define WW 1024
#define CH_HW (HH * WW)          // elements per (b,c) channel image
#define NCH 24                   // 8 batches * 3 channels

// k_log tiling
#define TS 32                    // output tile edge
#define H1 34                    // tile + halo 1 (3x3 stencil)
#define H2 36                    // tile + halo 2 (5x5 effective LoG)
#define S1 36                    // LDS row stride for halo-1 tiles
#define S2 40                    // LDS row stride for halo-2 tiles

// ---------------------------------------------------------------------------
// CDNA5 async global->LDS helpers (inline asm: portable across toolchains)
// ---------------------------------------------------------------------------
__device__ __forceinline__ void async_ld16(unsigned lds_off, const void* gaddr) {
    asm volatile("global_load_async_to_lds_b128 %0, %1, off"
                 :: "v"(lds_off), "v"(gaddr) : "memory");
}
__device__ __forceinline__ void async_ld4(unsigned lds_off, const void* gaddr) {
    asm volatile("global_load_async_to_lds_b32 %0, %1, off"
                 :: "v"(lds_off), "v"(gaddr) : "memory");
}
__device__ __forceinline__ void wait_async0() {
    asm volatile("s_wait_asynccnt 0" ::: "memory");
}
__device__ __forceinline__ void wait_async2() {
    asm volatile("s_wait_asynccnt 2" ::: "memory");
}

// ---------------------------------------------------------------------------
__device__ __forceinline__ double block_reduce(double v, double* red) {
    const int tid = threadIdx.x;
    red[tid] = v;
    __syncthreads();
    for (int s = blockDim.x >> 1; s > 0; s >>= 1) {
        if (tid < s) red[tid] += red[tid + s];
        __syncthreads();
    }
    double r = red[0];
    __syncthreads();
    return r;
}

// reflect101 (numpy 'reflect'): -1 -> 1, n -> n-2 (pad width <= 2 here)
__device__ __forceinline__ int refl(int i, int n) {
    i = (i < 0) ? -i : i;
    i = (i >= n) ? (2 * n - 2 - i) : i;
    return i;
}

// ---------------------------------------------------------------------------
// ws accumulator layout (doubles):
//  [0] sat | [1..24] sum I | [25..48] sumsq I | [49..72] sum I_D
//  [73..96] sumsq I_D | [97] sobel | [98] log
// ---------------------------------------------------------------------------
__global__ void k_zero(double* acc) {
    if (threadIdx.x < 128) acc[threadIdx.x] = 0.0;
}

// ---------------------------------------------------------------------------
// K1: streaming stats, one channel image per blockIdx.y.
// Double-buffered async DMA: 256 lanes x 16B per image per chunk.
// ---------------------------------------------------------------------------
__global__ void __launch_bounds__(256) k_stats(const float* __restrict__ I,
                                               const float* __restrict__ ID,
                                               double* __restrict__ acc) {
    const int tid = threadIdx.x;
    const int ch  = blockIdx.y;
    const size_t base = (size_t)ch * (size_t)CH_HW;

    __shared__ __align__(16) float sb[2][2][256 * 4];  // 16 KB staging
    __shared__ double red[256];

    unsigned loff[2][2];
#pragma unroll
    for (int b = 0; b < 2; ++b)
#pragma unroll
        for (int im = 0; im < 2; ++im)
            loff[b][im] = (unsigned)(size_t)(&sb[b][im][tid * 4]);

    const int nchunks = CH_HW / 1024;
    const int stride  = gridDim.x;

    double sI = 0.0, sI2 = 0.0, sD = 0.0, sD2 = 0.0, sat = 0.0;

    int c = blockIdx.x;
    if (c < nchunks) {
        async_ld16(loff[0][0], I  + base + (size_t)c * 1024 + tid * 4);
        async_ld16(loff[0][1], ID + base + (size_t)c * 1024 + tid * 4);
    }
    int buf = 0;
    for (; c < nchunks; c += stride) {
        const int cn = c + stride;
        if (cn < nchunks) {
            async_ld16(loff[buf ^ 1][0], I  + base + (size_t)cn * 1024 + tid * 4);
            async_ld16(loff[buf ^ 1][1], ID + base + (size_t)cn * 1024 + tid * 4);
            wait_async2();                // current buffer's 2 ops done (in-order)
        } else {
            wait_async0();
        }
        const float4 vI = *(const float4*)&sb[buf][0][tid * 4];
        const float4 vD = *(const float4*)&sb[buf][1][tid * 4];
        const float xs[4]  = {vI.x, vI.y, vI.z, vI.w};
        const float ds4[4] = {vD.x, vD.y, vD.z, vD.w};
#pragma unroll
        for (int k = 0; k < 4; ++k) {
            const float x = xs[k];
            const float r = fmaxf(-x, 0.0f) + fmaxf(x - 1.0f, 0.0f);
            sat += (double)(r * r);
            sI  += (double)x;
            sI2 += (double)x * (double)x;
            const float d = ds4[k];
            sD  += (double)d;
            sD2 += (double)d * (double)d;
        }
        buf ^= 1;
    }

    double t;
    t = block_reduce(sat, red); if (tid == 0) atomicAdd(&acc[0], t);
    t = block_reduce(sI,  red); if (tid == 0) atomicAdd(&acc[1 + ch], t);
    t = block_reduce(sI2, red); if (tid == 0) atomicAdd(&acc[25 + ch], t);
    t = block_reduce(sD,  red); if (tid == 0) atomicAdd(&acc[49 + ch], t);
    t = block_reduce(sD2, red); if (tid == 0) atomicAdd(&acc[73 + ch], t);
}

// ---------------------------------------------------------------------------
// K2: degenerate sobel on batch0/channel0 (edge columns identically zero).
// ---------------------------------------------------------------------------
__global__ void __launch_bounds__(256) k_sobel(const float* __restrict__ P,
                                               const float* __restrict__ T,
                                               double* __restrict__ acc) {
    __shared__ double red[256];
    double s = 0.0;
    const int n = HH * WW;
    for (int idx = blockIdx.x * blockDim.x + threadIdx.x; idx < n;
         idx += gridDim.x * blockDim.x) {
        const int y = idx >> 10;
        const int x = idx & (WW - 1);
        float v = 0.0f;
        if (x >= 1 && x <= WW - 2) {
            const int r = y * WW;
            const float dp = P[r + x - 1] - P[r + x + 1];
            const float dt = T[r + x - 1] - T[r + x + 1];
            v = fabsf(4.0f * dp - 4.0f * dt);
        }
        s += (double)v;
    }
    const double t = block_reduce(s, red);
    if (threadIdx.x == 0) atomicAdd(&acc[97], t);
}

// ---------------------------------------------------------------------------
// K3: log loss, LDS-tiled. 32x32 outputs/block, 256 threads (4 px/thread).
// Stage 1: async-DMA img[0,0] halo-2 tiles -> LDS, build gauss tiles
//          Ghat(g) = gauss(img)(refl(g)) at halo-1, lap values in registers.
//          (slots indexed by UNreflected coords, values at reflected coords:
//           this reproduces filt3(filt3(.)) nested reflect101 exactly)
// Stage 2: per batch, gray halo-1 tiles in LDS, 9 ds_reads/pixel gauss.
// ---------------------------------------------------------------------------
__global__ void __launch_bounds__(256) k_log(const float* __restrict__ I,
                                             const float* __restrict__ ID,
                                             double* __restrict__ acc) {
    __shared__ __align__(16) float bufA[H2 * S2];  // img P00 halo-2 tile
    __shared__ __align__(16) float bufB[H2 * S2];  // img T00 halo-2 tile
    __shared__ float gA[H1 * S1];                  // Ghat P  /  gray P
    __shared__ float gB[H1 * S1];                  // Ghat T  /  gray T
    __shared__ double red[256];

    const float Kg[3][3] = {{1.f/16.f, 2.f/16.f, 1.f/16.f},
                            {2.f/16.f, 4.f/16.f, 2.f/16.f},
                            {1.f/16.f, 2.f/16.f, 1.f/16.f}};

    const int tid = threadIdx.x;
    const int oy = blockIdx.y * TS;
    const int ox = blockIdx.x * TS;

    // ---- stage 1a: async load img[0,0] tiles (halo 2, reflect at borders)
    for (int s = tid; s < H2 * H2; s += 256) {
        const int r = s / H2, c = s % H2;
        const size_t off = (size_t)refl(oy - 2 + r, HH) * WW + refl(ox - 2 + c, WW);
        async_ld4((unsigned)(size_t)&bufA[r * S2 + c], I  + off);
        async_ld4((unsigned)(size_t)&bufB[r * S2 + c], ID + off);
    }
    wait_async0();
    __syncthreads();

    // ---- stage 1b: Ghat(g) = gauss(img)(refl(g)) on the halo-1 tile
    for (int s = tid; s < H1 * H1; s += 256) {
        const int r = s / H1, c = s % H1;
        const int ry = refl(oy - 1 + r, HH);   // valid coord whose gauss we want
        const int rx = refl(ox - 1 + c, WW);
        float sp = 0.0f, st = 0.0f;
#pragma unroll
        for (int dy = 0; dy < 3; ++dy) {
            const int lr = (ry + dy - 1) - (oy - 2);  // unreflected idx into halo-2
#pragma unroll
            for (int dx = 0; dx < 3; ++dx) {
                const int lc = (rx + dx - 1) - (ox - 2);
                const float k = Kg[dy][dx];
                sp += k * bufA[lr * S2 + lc];
                st += k * bufB[lr * S2 + lc];
            }
        }
        gA[r * S1 + c] = sp;
        gB[r * S1 + c] = st;
    }
    __syncthreads();

    // ---- stage 1c: per-pixel LoG values into registers
    const int px  = tid & 31;
    const int py0 = tid >> 5;    // 0..7, pixels at rows py0 + 8k
    float lapP[4], lapT[4];
#pragma unroll
    for (int k = 0; k < 4; ++k) {
        const int i0 = (py0 + 8 * k + 1) * S1 + (px + 1);  // +1: tile origin o-1
        lapP[k] = 4.0f * gA[i0] - gA[i0 - S1] - gA[i0 + S1] - gA[i0 - 1] - gA[i0 + 1];
        lapT[k] = 4.0f * gB[i0] - gB[i0 - S1] - gB[i0 + S1] - gB[i0 - 1] - gB[i0 + 1];
    }

    // ---- stage 2: per batch, gray tiles then gauss(gray)
    double s = 0.0;
    for (int b = 0; b < 8; ++b) {
        __syncthreads();                       // previous readers of gA/gB done
        const size_t cb = (size_t)(3 * b) * (size_t)CH_HW;
        for (int t = tid; t < H1 * H1; t += 256) {
            const int r = t / H1, c = t % H1;
            const size_t off = cb + (size_t)refl(oy - 1 + r, HH) * WW
                                  + refl(ox - 1 + c, WW);
            gA[r * S1 + c] = (I [off] + I [off + CH_HW] + I [off + 2 * CH_HW]) / 3.0f;
            gB[r * S1 + c] = (ID[off] + ID[off + CH_HW] + ID[off + 2 * CH_HW]) / 3.0f;
        }
        __syncthreads();
#pragma unroll
        for (int k = 0; k < 4; ++k) {
            const int ly = py0 + 8 * k + 1;
            const int lx = px + 1;
            float gp = 0.0f, gt = 0.0f;
#pragma unroll
            for (int dy = 0; dy < 3; ++dy)
#pragma unroll
                for (int dx = 0; dx < 3; ++dx) {
                    const float kk = Kg[dy][dx];
                    const int ii = (ly + dy - 1) * S1 + (lx + dx - 1);
                    gp += kk * gA[ii];
                    gt += kk * gB[ii];
                }
            s += (double)fabsf(gp * lapP[k] - gt * lapT[k]);
        }
    }
    __syncthreads();
    const double t = block_reduce(s, red);
    if (tid == 0) atomicAdd(&acc[98], t);
}

// ---------------------------------------------------------------------------
// K4: combine partial sums into the scalar loss.
// ---------------------------------------------------------------------------
__global__ void k_final(const double* __restrict__ acc, float* __restrict__ out) {
    if (threadIdx.x == 0 && blockIdx.x == 0) {
        const double HW = (double)CH_HW;
        const double L_sat = acc[0] / ((double)NCH * HW);
        double Lint = 0.0, Lsp = 0.0;
        for (int c = 0; c < NCH; ++c) {
            const double sumI = acc[1 + c],  sqI = acc[25 + c];
            const double sumD = acc[49 + c], sqD = acc[73 + c];
            const double mI = sumI / HW;
            double vI = (sqI - sumI * sumI / HW) / (HW - 1.0);
            double vD = (sqD - sumD * sumD / HW) / (HW - 1.0);
            vI = vI > 0.0 ? vI : 0.0;
            vD = vD > 0.0 ? vD : 0.0;
            Lint += (mI - 0.5) * (mI - 0.5);
            const double d = sqrt(vI) - sqrt(vD);
            Lsp += d * d;
        }
        Lint /= (double)NCH;
        Lsp  /= (double)NCH;
        const double Lsob = acc[97] / HW;
        const double Llog = acc[98] / (8.0 * HW);
        out[0] = (float)(L_sat + Lsp + Lsob + Lint + Llog);
    }
}

// ---------------------------------------------------------------------------
extern "C" void kernel_launch(void* const* d_in, const int* in_sizes, int n_in,
                              void* d_out, int out_size, void* d_ws, size_t ws_size,
                              hipStream_t stream) {
    (void)in_sizes; (void)n_in; (void)out_size; (void)ws_size;
    const float* I_D = (const float*)d_in[0];   // setup_inputs order: I_D, I
    const float* I   = (const float*)d_in[1];
    double* acc = (double*)d_ws;                // 128 doubles of scratch
    float* out = (float*)d_out;

    k_zero<<<1, 128, 0, stream>>>(acc);

    dim3 gs(64, NCH);
    k_stats<<<gs, 256, 0, stream>>>(I, I_D, acc);

    k_sobel<<<1024, 256, 0, stream>>>(I, I_D, acc);

    dim3 gl(WW / TS, HH / TS);                  // 32 x 32 tiles
    k_log<<<gl, 256, 0, stream>>>(I, I_D, acc);

    k_final<<<1, 32, 0, stream>>>(acc, out);
}